// GCN_32607391711820
// MI455X (gfx1250) — compile-verified
//
#include <hip/hip_runtime.h>

#define N_NODES 100000
#define N_EDGES 1600000
#define D_FEAT  128

typedef float v2f __attribute__((ext_vector_type(2)));
typedef float v8f __attribute__((ext_vector_type(8)));

// ---------------- utility kernels ----------------

__global__ void gcn_zero_f32x4(float4* __restrict__ p, int n4) {
    int i = blockIdx.x * blockDim.x + threadIdx.x;
    if (i < n4) p[i] = make_float4(0.f, 0.f, 0.f, 0.f);
}

__global__ void gcn_degree_count(const int* __restrict__ src,
                                 const int* __restrict__ dst,
                                 float* __restrict__ deg_out,
                                 float* __restrict__ deg_in, int nE) {
    int e = blockIdx.x * blockDim.x + threadIdx.x;
    if (e < nE) {
        (void)__hip_atomic_fetch_add(&deg_out[src[e]], 1.0f,
                                     __ATOMIC_RELAXED, __HIP_MEMORY_SCOPE_AGENT);
        (void)__hip_atomic_fetch_add(&deg_in[dst[e]], 1.0f,
                                     __ATOMIC_RELAXED, __HIP_MEMORY_SCOPE_AGENT);
    }
}

__global__ void gcn_inv_sqrt(float* __restrict__ p, int n) {
    int i = blockIdx.x * blockDim.x + threadIdx.x;
    if (i < n) {
        float d = p[i];
        p[i] = (d > 0.f) ? rsqrtf(d) : 0.f;
    }
}

// ---------------- edge scatter (segment_sum) ----------------
// One wave32 per edge: lane L owns floats [4L, 4L+4) of the 128-float row.
// Coalesced float4 gather of h[src], fp32 global atomic adds into agg[dst].
__global__ void gcn_scatter_edges(const float* __restrict__ h,
                                  const int* __restrict__ src,
                                  const int* __restrict__ dst,
                                  const float* __restrict__ isq_out,
                                  float* __restrict__ agg, int nE) {
    int e = blockIdx.x * (blockDim.x >> 5) + (threadIdx.x >> 5);
    if (e >= nE) return;
    int lane = threadIdx.x & 31;
    int s = src[e];
    int d = dst[e];
    float w = isq_out[s];
    const float4* hp = (const float4*)(h + (size_t)s * D_FEAT) + lane;
    float4 v = *hp;  // global_load_b128
    float* ap = agg + (size_t)d * D_FEAT + lane * 4;
    float vals[4] = {v.x * w, v.y * w, v.z * w, v.w * w};
#pragma unroll
    for (int i = 0; i < 4; ++i)
        (void)__hip_atomic_fetch_add(ap + i, vals[i],
                                     __ATOMIC_RELAXED, __HIP_MEMORY_SCOPE_AGENT);
}

// ---------------- WMMA GEMM: out = [relu]( (agg * isq_in) @ W + b ) ----------
// One block = 16 output rows. Wave w owns output columns [16w, 16w+16).
// blockDim.x = 32 * (NOUT/16). K = 128 fixed, fp32 WMMA 16x16x4.
// NOUT/RELU are compile-time so all W loads / out stores use immediate offsets.
template <int NOUT, bool RELU>
__global__ void gcn_gemm_wmma(const float* __restrict__ A,
                              const float* __restrict__ isq_in,
                              const float* __restrict__ W,
                              const float* __restrict__ bias,
                              float* __restrict__ out) {
    __shared__ float sA[16][132];   // pad 132: 64-bank LDS, stride 128 would conflict
    __shared__ float sIsq[16];

    const int tid  = threadIdx.x;
    const int row0 = blockIdx.x * 16;

    if (tid < 16) sIsq[tid] = isq_in[row0 + tid];
    __syncthreads();

#pragma unroll
    for (int e = tid; e < 16 * D_FEAT; e += 32 * (NOUT / 16)) {
        int r = e >> 7;
        int c = e & 127;
        sA[r][c] = A[(size_t)(row0 + r) * D_FEAT + c] * sIsq[r];
    }
    __syncthreads();

    const int lane    = tid & 31;
    const int colbase = (tid >> 5) * 16;
    const int col     = lane & 15;
    const int kh      = (lane >> 4) << 1;  // lanes 16-31 hold K+2,K+3 (ISA 7.12.2)
    const int rowA    = lane & 15;

    // Uniform per-lane base pointer; row offsets become compile-time immediates.
    const float* wp = W + (size_t)kh * NOUT + colbase + col;

    v8f acc = {};
#pragma unroll
    for (int k0 = 0; k0 < D_FEAT; k0 += 4) {
        v2f a, b;
        a[0] = sA[rowA][k0 + kh];
        a[1] = sA[rowA][k0 + kh + 1];
        b[0] = wp[k0 * NOUT];            // W[(k0+kh)  ][col] : imm offset
        b[1] = wp[k0 * NOUT + NOUT];     // W[(k0+kh+1)][col] : imm offset
        acc = __builtin_amdgcn_wmma_f32_16x16x4_f32(
            /*neg_a=*/false, a, /*neg_b=*/false, b,
            /*c_mod=*/(short)0, acc, /*reuse_a=*/false, /*reuse_b=*/false);
    }

    // D layout: VGPR r -> M = r (+8 for lanes 16-31), N = lane&15
    float bb = bias[colbase + col];
    int mrow = row0 + ((lane >> 4) << 3);
    float* op = out + (size_t)mrow * NOUT + colbase + col;
#pragma unroll
    for (int r = 0; r < 8; ++r) {
        float v = acc[r] + bb;
        if (RELU) v = fmaxf(v, 0.f);
        op[r * NOUT] = v;                // imm offsets
    }
}

// ---------------- launch ----------------

extern "C" void kernel_launch(void* const* d_in, const int* in_sizes, int n_in,
                              void* d_out, int out_size, void* d_ws, size_t ws_size,
                              hipStream_t stream) {
    const float* x  = (const float*)d_in[0];
    const float* W1 = (const float*)d_in[1];
    const float* b1 = (const float*)d_in[2];
    const float* W2 = (const float*)d_in[3];
    const float* b2 = (const float*)d_in[4];
    const float* W3 = (const float*)d_in[5];
    const float* b3 = (const float*)d_in[6];
    const int*  src = (const int*)d_in[7];
    const int*  dst = (const int*)d_in[8];
    float* out = (float*)d_out;

    float* isq_out = (float*)d_ws;                         // [N]
    float* isq_in  = isq_out + N_NODES;                    // [N]
    float* agg     = isq_in + N_NODES;                     // [N,128]
    float* h       = agg + (size_t)N_NODES * D_FEAT;       // [N,128]

    const int aggN4 = (N_NODES * D_FEAT) / 4;              // 3,200,000
    const int edgeBlocks = N_EDGES / 8;                    // 8 waves/block, 1 edge/wave

    // degree normalization
    {
        int n4 = (2 * N_NODES) / 4;
        gcn_zero_f32x4<<<(n4 + 255) / 256, 256, 0, stream>>>((float4*)isq_out, n4);
    }
    gcn_degree_count<<<N_EDGES / 256, 256, 0, stream>>>(src, dst, isq_out, isq_in, N_EDGES);
    gcn_inv_sqrt<<<(2 * N_NODES + 255) / 256, 256, 0, stream>>>(isq_out, 2 * N_NODES);

    // layer 1: x -> h
    gcn_zero_f32x4<<<(aggN4 + 255) / 256, 256, 0, stream>>>((float4*)agg, aggN4);
    gcn_scatter_edges<<<edgeBlocks, 256, 0, stream>>>(x, src, dst, isq_out, agg, N_EDGES);
    gcn_gemm_wmma<128, true><<<N_NODES / 16, 256, 0, stream>>>(agg, isq_in, W1, b1, h);

    // layer 2: h -> h
    gcn_zero_f32x4<<<(aggN4 + 255) / 256, 256, 0, stream>>>((float4*)agg, aggN4);
    gcn_scatter_edges<<<edgeBlocks, 256, 0, stream>>>(h, src, dst, isq_out, agg, N_EDGES);
    gcn_gemm_wmma<128, true><<<N_NODES / 16, 256, 0, stream>>>(agg, isq_in, W2, b2, h);

    // layer 3: h -> out (64 cols, 4 waves/block, no relu)
    gcn_zero_f32x4<<<(aggN4 + 255) / 256, 256, 0, stream>>>((float4*)agg, aggN4);
    gcn_scatter_edges<<<edgeBlocks, 256, 0, stream>>>(h, src, dst, isq_out, agg, N_EDGES);
    gcn_gemm_wmma<64, false><<<N_NODES / 16, 128, 0, stream>>>(agg, isq_in, W3, b3, out);
}